// Embedding_45037027066139
// MI455X (gfx1250) — compile-verified
//
#include <hip/hip_runtime.h>
#include <stdint.h>

// Embedding lookup: y[r, :] = W[indices[r], :], r in [0, 8192), DIM=128 (512 B/row).
// One wave32 per row; each lane DMAs 16 B via the CDNA5 async global<->LDS path.

#define EMB_DIM        128
#define ROW_BYTES      (EMB_DIM * 4)   // 512 bytes per row
#define WAVES_PER_BLK  8
#define BLOCK_THREADS  (WAVES_PER_BLK * 32)

__global__ __launch_bounds__(BLOCK_THREADS)
void emb_gather_async_kernel(const int* __restrict__ indices,
                             const float* __restrict__ W,
                             float* __restrict__ out,
                             unsigned n_rows)
{
    // Per-wave staging buffer in LDS: 8 waves * 512 B = 4 KB.
    __shared__ __attribute__((aligned(16))) char smem[WAVES_PER_BLK * ROW_BYTES];

    const unsigned lane       = threadIdx.x & 31u;
    const unsigned wave_in_bk = threadIdx.x >> 5;
    const unsigned row        = blockIdx.x * WAVES_PER_BLK + wave_in_bk;
    if (row >= n_rows) return;   // grid is sized exact; guard is defensive only

    // Embedding row to fetch (wave-uniform value, per-lane load is fine).
    const int e = indices[row];

    // LDS byte offset for this lane: low 32 bits of the flat shared-mem address
    // are the LDS offset (aperture bits live in addr[63:32]).
    const unsigned lds_off =
        (unsigned)(uintptr_t)&smem[wave_in_bk * ROW_BYTES + lane * 16u];

    const char* src = (const char*)(W + (size_t)e * EMB_DIM) + lane * 16u;
    char*       dst = (char*)(out + (size_t)row * EMB_DIM)   + lane * 16u;

    // HBM -> LDS, async DMA, 16 B per lane = full 512 B row per wave.
    asm volatile("global_load_async_to_lds_b128 %0, %1, off"
                 :: "v"(lds_off), "v"(src)
                 : "memory");

    // The dependent store reads LDS written by the async load; async ops are
    // not LDS-ordered with each other, so drain ASYNCcnt first.
    asm volatile("s_wait_asynccnt 0" ::: "memory");

    // LDS -> HBM, async DMA. S_ENDPGM performs an implicit wait-idle, so the
    // store is guaranteed complete before the wave retires.
    asm volatile("global_store_async_from_lds_b128 %0, %1, off"
                 :: "v"(dst), "v"(lds_off)
                 : "memory");
}

extern "C" void kernel_launch(void* const* d_in, const int* in_sizes, int n_in,
                              void* d_out, int out_size, void* d_ws, size_t ws_size,
                              hipStream_t stream)
{
    const int*   indices = (const int*)d_in[0];   // [4, 2048] int32
    const float* W       = (const float*)d_in[1]; // [50257, 128] f32
    float*       out     = (float*)d_out;         // [4, 2048, 128] f32

    const unsigned n_rows = (unsigned)in_sizes[0];                 // 8192
    const unsigned blocks = (n_rows + WAVES_PER_BLK - 1) / WAVES_PER_BLK; // 1024

    emb_gather_async_kernel<<<blocks, BLOCK_THREADS, 0, stream>>>(indices, W, out, n_rows);
}